// VariantCoeLinear1d_83064667505016
// MI455X (gfx1250) — compile-verified
//
#include <hip/hip_runtime.h>

#define NT 512          // threads per block (16 wave32)
#define NE 8            // elements per thread
#define NN 4096         // grid points per row
static_assert(NT * NE == NN, "layout");

#define B12 (0.5f / 12.0f)      // BETA/12
#define LAM 0.4096f             // DT/DX = 0.001 / (10/4096)

// ---- CDNA5 async LDS->global store path (probe via __has_builtin) ----
#if defined(__has_builtin)
#if __has_builtin(__builtin_amdgcn_global_store_async_from_lds_b128)
#define HAS_ASYNC_STORE 1
#endif
#if __has_builtin(__builtin_amdgcn_s_wait_asynccnt)
#define HAS_WAIT_ASYNC 1
#endif
#endif

#if defined(HAS_ASYNC_STORE)
// In-loop reuse guard MUST be a compiler memory barrier so the staging-buffer
// LDS stores cannot be hoisted above it -> use inline asm with "memory".
#define WAIT_ASYNCCNT_BARRIER(n) \
  asm volatile("s_wait_asynccnt %0" ::"i"(n) : "memory")
#if defined(HAS_WAIT_ASYNC)
#define WAIT_ASYNCCNT_FINAL(n) __builtin_amdgcn_s_wait_asynccnt(n)
#else
#define WAIT_ASYNCCNT_FINAL(n) WAIT_ASYNCCNT_BARRIER(n)
#endif
#define WAIT_DSCNT0() asm volatile("s_wait_dscnt 0" ::: "memory")
// builtin signature (from clang diagnostic): (int4 AS1*, int4 AS3*, Ii, Ii)
typedef int v4i __attribute__((vector_size(4 * sizeof(int))));
#define GPTR(p) ((__attribute__((address_space(1))) v4i*)(p))
#define LPTR(p) ((__attribute__((address_space(3))) v4i*)(p))
#endif

__device__ __forceinline__ float flux_f(float u) {
  float u2 = u * u;
  return 0.5f * u * (3.0f - u2) +
         B12 * u2 * (0.75f - 2.0f * u + 1.5f * u2 - 0.25f * u2 * u2);
}

__device__ __forceinline__ float adflux_f(float u) {
  float u2 = u * u;
  float u3 = u2 * u;
  return fabsf(1.5f - 1.5f * u2 +
               B12 * (1.5f * u - 6.0f * u2 + 6.0f * u3 - 1.5f * u3 * u2));
}

extern "C" __global__ void __launch_bounds__(NT)
lf1d_march(const float* __restrict__ init, float* __restrict__ out,
           int nbatch, int stepnum) {
  __shared__ float s_first[NT];     // first element of each thread's chunk
  __shared__ float s_last[NT];      // last  element of each thread's chunk
  __shared__ float s_obuf[2][NN];   // ping-pong staging for async output DMA

  const int t = threadIdx.x;
  const int b = blockIdx.x;
  const int base = t * NE;

  // ---- load init row chunk into registers; emit out[0] = init ----
  const float* rowin = init + (size_t)b * NN + base;
  float4 v0 = *(const float4*)(rowin);
  float4 v1 = *(const float4*)(rowin + 4);
  float u[NE] = {v0.x, v0.y, v0.z, v0.w, v1.x, v1.y, v1.z, v1.w};
  {
    float* o0 = out + (size_t)b * NN + base;
    *(float4*)(o0) = v0;
    *(float4*)(o0 + 4) = v1;
  }

  for (int s = 1; s < stepnum; ++s) {
    // halo exchange (only 2 floats cross thread boundaries)
    s_first[t] = u[0];
    s_last[t] = u[NE - 1];
    __syncthreads();
    float ul = (t > 0) ? s_last[t - 1] : 0.0f;        // u[base-1]
    float ur = (t < NT - 1) ? s_first[t + 1] : 0.0f;  // u[base+NE]

    // local window uu[0..NE+1] = u[base-1 .. base+NE]
    float uu[NE + 2];
    uu[0] = ul;
#pragma unroll
    for (int k = 0; k < NE; ++k) uu[k + 1] = u[k];
    uu[NE + 1] = ur;

    float f[NE + 2], a[NE + 2];
#pragma unroll
    for (int k = 0; k < NE + 2; ++k) {
      f[k] = flux_f(uu[k]);
      a[k] = adflux_f(uu[k]);
    }

    // fh[i] = Lax-Friedrichs flux at global interface (base-1+i)
    float fh[NE + 1];
#pragma unroll
    for (int i = 0; i < NE + 1; ++i)
      fh[i] = 0.5f * (f[i] + f[i + 1]) -
              0.5f * fmaxf(a[i], a[i + 1]) * (uu[i + 1] - uu[i]);

    float un[NE];
#pragma unroll
    for (int j = 0; j < NE; ++j)
      un[j] = uu[j + 1] - LAM * (fh[j + 1] - fh[j]);
    if (t == 0) un[0] = un[1];                // u_new[0]   = u_new[1]
    if (t == NT - 1) un[NE - 1] = un[NE - 2]; // u_new[N-1] = u_new[N-2]

#pragma unroll
    for (int j = 0; j < NE; ++j) u[j] = un[j];

    float4 w0 = make_float4(un[0], un[1], un[2], un[3]);
    float4 w1 = make_float4(un[4], un[5], un[6], un[7]);
    float* orow = out + ((size_t)s * nbatch + b) * NN + base;

#if defined(HAS_ASYNC_STORE)
    // reuse guard: 2 async stores/wave/step, in-order completion; allow the
    // most recent step's 2 in flight, guarantee the (s-2) buffer drained.
    WAIT_ASYNCCNT_BARRIER(2);
    float* ob = &s_obuf[s & 1][base];
    *(float4*)(ob) = w0;
    *(float4*)(ob + 4) = w1;
    WAIT_DSCNT0();  // async engine's LDS reads are unordered vs DS stores
    __builtin_amdgcn_global_store_async_from_lds_b128(GPTR(orow), LPTR(ob), 0, 0);
    __builtin_amdgcn_global_store_async_from_lds_b128(GPTR(orow), LPTR(ob), 16, 0);
#else
    *(float4*)(orow) = w0;
    *(float4*)(orow + 4) = w1;
#endif
    __syncthreads();
  }

#if defined(HAS_ASYNC_STORE)
  WAIT_ASYNCCNT_FINAL(0);
#endif
}

extern "C" void kernel_launch(void* const* d_in, const int* in_sizes, int n_in,
                              void* d_out, int out_size, void* d_ws,
                              size_t ws_size, hipStream_t stream) {
  const float* init = (const float*)d_in[0];
  float* out = (float*)d_out;
  const int total = in_sizes[0];         // B * N
  const int nbatch = total / NN;         // 256
  const int stepnum = out_size / total;  // 100
  lf1d_march<<<dim3(nbatch), dim3(NT), 0, stream>>>(init, out, nbatch, stepnum);
}